// ToyMoEKernel_25151328485597
// MI455X (gfx1250) — compile-verified
//
#include <hip/hip_runtime.h>

// ToyMoE on gfx1250: fused gate_up GEMM + SwiGLU, then down GEMM.
// fp32 path via V_WMMA_F32_16X16X4_F32 (memory-bound problem; fp32 WMMA is
// far above the ~14 TFLOPS needed to saturate 23.3 TB/s HBM).
// Global->LDS staging uses gfx1250 async-to-LDS (ASYNCcnt) with double
// buffering so the next K-slab streams from HBM while WMMAs run.

typedef __attribute__((ext_vector_type(2))) float v2f;
typedef __attribute__((ext_vector_type(8))) float v8f;
typedef __attribute__((ext_vector_type(4))) int v4i;

typedef __attribute__((address_space(1))) v4i* g_v4i_ptr;  // global
typedef __attribute__((address_space(3))) v4i* l_v4i_ptr;  // LDS

#define DEV __device__ __forceinline__

#if __has_builtin(__builtin_amdgcn_global_load_async_to_lds_b128)
#define MOE_ASYNC_LDS 1
#else
#define MOE_ASYNC_LDS 0
#endif

DEV void cp16(float* lds, const float* glob) {
#if MOE_ASYNC_LDS
  __builtin_amdgcn_global_load_async_to_lds_b128(
      (g_v4i_ptr)(void*)glob, (l_v4i_ptr)(void*)lds,
      /*offset=*/0, /*cpol=*/0);
#else
  *(float4*)lds = *(const float4*)glob;
#endif
}

DEV void cp_wait() {
#if MOE_ASYNC_LDS
#if __has_builtin(__builtin_amdgcn_s_wait_asynccnt)
  __builtin_amdgcn_s_wait_asynccnt(0);
#else
  asm volatile("s_wait_asynccnt 0" ::: "memory");
#endif
#endif
}

DEV v8f wmma_f32(v2f a, v2f b, v8f c) {
  return __builtin_amdgcn_wmma_f32_16x16x4_f32(false, a, false, b,
                                               (short)0, c, false, false);
}

// ---------------------------------------------------------------------------
// Kernel 1: gu = x @ W1 + b1 ; hidden = silu(gate) * up
// Tile: 128 (T) x 128 (I), computing gate cols [n,n+128) AND up cols
// [I+n, I+n+128) in one pass so SwiGLU fuses into the epilogue.
// ---------------------------------------------------------------------------
__global__ __launch_bounds__(256) void moe_gate_up_silu_kernel(
    const float* __restrict__ x, const float* __restrict__ w1,
    const float* __restrict__ b1, float* __restrict__ hidden) {
  constexpr int T = 1024, H = 2048, I = 4096;
  constexpr int KT = 32;
  constexpr int APAD = 36;   // 32 + 4 pad, keeps 16B alignment, spreads banks
  constexpr int BPAD = 132;  // 128 + 4 pad

  __shared__ float sA[2][128 * APAD];
  __shared__ float sBg[2][KT * BPAD];
  __shared__ float sBu[2][KT * BPAD];

  const int e = blockIdx.z;
  const int mBase = blockIdx.x * 128;  // over T
  const int nBase = blockIdx.y * 128;  // over I (gate side)

  const float* A = x + (size_t)e * T * H;
  const float* W = w1 + (size_t)e * H * (2 * I);
  const float* bias = b1 + (size_t)e * (2 * I);
  float* hout = hidden + (size_t)e * T * I;

  const int tid = threadIdx.x;
  const int lane = tid & 31;
  const int wave = tid >> 5;
  const int mW = (wave >> 2) * 64;  // wave tile origin inside 128x128
  const int nW = (wave & 3) * 32;
  const int mL = lane & 15;
  const int kHi = (lane >> 4) << 1;  // A/B frag: upper half-wave holds K+2,K+3

  v8f accG[4][2] = {};
  v8f accU[4][2] = {};

  auto load_tiles = [&](int buf, int k0) {
    // A tile: 128 rows x 32 cols (row-major in x), 8 float4 per row
    for (int i = tid; i < 128 * 8; i += 256) {
      const int r = i >> 3, c = (i & 7) << 2;
      cp16(&sA[buf][r * APAD + c], &A[(size_t)(mBase + r) * H + (k0 + c)]);
    }
    // B tiles: 32 K-rows x 128 cols; gate at nBase, up at I + nBase
    for (int i = tid; i < 32 * 32; i += 256) {
      const int r = i >> 5, c = (i & 31) << 2;
      const float* src = &W[(size_t)(k0 + r) * (2 * I) + nBase + c];
      cp16(&sBg[buf][r * BPAD + c], src);
      cp16(&sBu[buf][r * BPAD + c], src + I);
    }
  };

  load_tiles(0, 0);

  for (int k0 = 0; k0 < H; k0 += KT) {
    const int cur = (k0 / KT) & 1;
    cp_wait();        // this wave's async loads for `cur` have landed in LDS
    __syncthreads();  // every wave's have; prev compute on cur^1 is finished
    if (k0 + KT < H) load_tiles(cur ^ 1, k0 + KT);  // overlaps compute below

    const float* cA = sA[cur];
    const float* cBg = sBg[cur];
    const float* cBu = sBu[cur];

#pragma unroll
    for (int kk = 0; kk < KT; kk += 4) {
      const int kb = kk + kHi;
      v2f a[4], bg[2], bu[2];
#pragma unroll
      for (int mt = 0; mt < 4; ++mt) {
        const float* p = &cA[(mW + mt * 16 + mL) * APAD + kb];
        a[mt][0] = p[0];
        a[mt][1] = p[1];
      }
#pragma unroll
      for (int nt = 0; nt < 2; ++nt) {
        const int c = nW + nt * 16 + mL;
        bg[nt][0] = cBg[kb * BPAD + c];
        bg[nt][1] = cBg[(kb + 1) * BPAD + c];
        bu[nt][0] = cBu[kb * BPAD + c];
        bu[nt][1] = cBu[(kb + 1) * BPAD + c];
      }
#pragma unroll
      for (int mt = 0; mt < 4; ++mt)
#pragma unroll
        for (int nt = 0; nt < 2; ++nt) {
          accG[mt][nt] = wmma_f32(a[mt], bg[nt], accG[mt][nt]);
          accU[mt][nt] = wmma_f32(a[mt], bu[nt], accU[mt][nt]);
        }
    }
  }

  // Epilogue: bias + SwiGLU, write hidden.  C layout: VGPR r holds M=r for
  // lanes 0-15 and M=r+8 for lanes 16-31; N = lane & 15.
  const int rOff = (lane >> 4) << 3;
#pragma unroll
  for (int mt = 0; mt < 4; ++mt) {
#pragma unroll
    for (int nt = 0; nt < 2; ++nt) {
      const int col = nBase + nW + nt * 16 + mL;
      const float bGate = bias[col];
      const float bUp = bias[I + col];
#pragma unroll
      for (int r = 0; r < 8; ++r) {
        const int row = mBase + mW + mt * 16 + rOff + r;
        const float g = accG[mt][nt][r] + bGate;
        const float u = accU[mt][nt][r] + bUp;
        const float h = (g / (1.0f + __expf(-g))) * u;  // silu(g) * u
        hout[(size_t)row * I + col] = h;
      }
    }
  }
}

// ---------------------------------------------------------------------------
// Kernel 2: out = hidden @ W2 + b2
// ---------------------------------------------------------------------------
__global__ __launch_bounds__(256) void moe_down_kernel(
    const float* __restrict__ hidden, const float* __restrict__ w2,
    const float* __restrict__ b2, float* __restrict__ out) {
  constexpr int T = 1024, H = 2048, I = 4096;
  constexpr int KT = 32;
  constexpr int APAD = 36;
  constexpr int BPAD = 132;

  __shared__ float sA[2][128 * APAD];
  __shared__ float sB[2][KT * BPAD];

  const int e = blockIdx.z;
  const int mBase = blockIdx.x * 128;  // over T
  const int nBase = blockIdx.y * 128;  // over H

  const float* A = hidden + (size_t)e * T * I;
  const float* W = w2 + (size_t)e * I * H;
  const float* bias = b2 + (size_t)e * H;
  float* O = out + (size_t)e * T * H;

  const int tid = threadIdx.x;
  const int lane = tid & 31;
  const int wave = tid >> 5;
  const int mW = (wave >> 2) * 64;
  const int nW = (wave & 3) * 32;
  const int mL = lane & 15;
  const int kHi = (lane >> 4) << 1;

  v8f acc[4][2] = {};

  auto load_tiles = [&](int buf, int k0) {
    for (int i = tid; i < 128 * 8; i += 256) {
      const int r = i >> 3, c = (i & 7) << 2;
      cp16(&sA[buf][r * APAD + c], &A[(size_t)(mBase + r) * I + (k0 + c)]);
    }
    for (int i = tid; i < 32 * 32; i += 256) {
      const int r = i >> 5, c = (i & 31) << 2;
      cp16(&sB[buf][r * BPAD + c], &W[(size_t)(k0 + r) * H + nBase + c]);
    }
  };

  load_tiles(0, 0);

  for (int k0 = 0; k0 < I; k0 += KT) {
    const int cur = (k0 / KT) & 1;
    cp_wait();
    __syncthreads();
    if (k0 + KT < I) load_tiles(cur ^ 1, k0 + KT);

    const float* cA = sA[cur];
    const float* cB = sB[cur];

#pragma unroll
    for (int kk = 0; kk < KT; kk += 4) {
      const int kb = kk + kHi;
      v2f a[4], b[2];
#pragma unroll
      for (int mt = 0; mt < 4; ++mt) {
        const float* p = &cA[(mW + mt * 16 + mL) * APAD + kb];
        a[mt][0] = p[0];
        a[mt][1] = p[1];
      }
#pragma unroll
      for (int nt = 0; nt < 2; ++nt) {
        const int c = nW + nt * 16 + mL;
        b[nt][0] = cB[kb * BPAD + c];
        b[nt][1] = cB[(kb + 1) * BPAD + c];
      }
#pragma unroll
      for (int mt = 0; mt < 4; ++mt)
#pragma unroll
        for (int nt = 0; nt < 2; ++nt)
          acc[mt][nt] = wmma_f32(a[mt], b[nt], acc[mt][nt]);
    }
  }

  const int rOff = (lane >> 4) << 3;
#pragma unroll
  for (int mt = 0; mt < 4; ++mt) {
#pragma unroll
    for (int nt = 0; nt < 2; ++nt) {
      const int col = nBase + nW + nt * 16 + mL;
      const float bD = bias[col];
#pragma unroll
      for (int r = 0; r < 8; ++r) {
        const int row = mBase + mW + mt * 16 + rOff + r;
        O[(size_t)row * H + col] = acc[mt][nt][r] + bD;
      }
    }
  }
}

extern "C" void kernel_launch(void* const* d_in, const int* in_sizes, int n_in,
                              void* d_out, int out_size, void* d_ws, size_t ws_size,
                              hipStream_t stream) {
  (void)in_sizes; (void)n_in; (void)out_size; (void)ws_size;
  constexpr int E = 8, T = 1024, H = 2048, I = 4096;

  const float* x   = (const float*)d_in[0];  // [E,T,H]
  const float* w1  = (const float*)d_in[1];  // [E,H,2I]
  const float* b1  = (const float*)d_in[2];  // [E,2I]
  const float* w2  = (const float*)d_in[3];  // [E,I,H]
  const float* b2  = (const float*)d_in[4];  // [E,H]
  float* out = (float*)d_out;                // [E,T,H]
  float* hidden = (float*)d_ws;              // [E,T,I] scratch, 128 MB

  dim3 g1(T / 128, I / 128, E);  // 8 x 32 x 8
  dim3 g2(T / 128, H / 128, E);  // 8 x 16 x 8
  moe_gate_up_silu_kernel<<<g1, 256, 0, stream>>>(x, w1, b1, hidden);
  moe_down_kernel<<<g2, 256, 0, stream>>>(hidden, w2, b2, out);
}